// AggXPredictor_19963007991854
// MI455X (gfx1250) — compile-verified
//
#include <hip/hip_runtime.h>
#include <stdint.h>

// AggXPredictor fuzzy min/mean/argmax pipeline for MI455X (gfx1250).
// Compute-bound on VALU/LDS (min/argmax semiring -> WMMA inapplicable);
// w (256KB) staged once per WGP into the 320KB LDS via CDNA5 async-to-LDS
// (ASYNCcnt) DMA, then two bank-conflict-free LDS passes with 4-row batching.

#define B_ROWS    2048
#define IN_DIM    256
#define OUT_DIM   256
#define ROW_STRIDE 260              // 256 + 4 dword pad: rows 16B-aligned, column reads conflict-free
#define RPG       4                 // batch rows amortized per w pass
#define SUBS      2                 // two 256-thread sub-blocks per 512-thread block
#define NTHREADS  512
#define ROWS_PER_BLOCK (RPG * SUBS) // 8
#define NBLOCKS   (B_ROWS / ROWS_PER_BLOCK) // 256
#define W_CHUNKS  (IN_DIM * OUT_DIM / 4)    // 16384 float4 chunks
#define CHUNKS_PER_THREAD (W_CHUNKS / NTHREADS) // 32, uniform

__device__ __forceinline__ uint32_t lds_lo32(const void* p) {
  // Generic pointers to LDS carry the byte offset in the low 32 bits
  // (aperture scheme); that is what VDST of the
  // async-to-LDS instruction expects.
  return (uint32_t)(uintptr_t)p;
}

__device__ __forceinline__ void async_ld_b128(uint32_t lds_off, const void* gptr) {
  asm volatile("global_load_async_to_lds_b128 %0, %1, off"
               :: "v"(lds_off), "v"(gptr) : "memory");
}
__device__ __forceinline__ void async_ld_b32(uint32_t lds_off, const void* gptr) {
  asm volatile("global_load_async_to_lds_b32 %0, %1, off"
               :: "v"(lds_off), "v"(gptr) : "memory");
}
__device__ __forceinline__ void wait_async_zero() {
  asm volatile("s_wait_asynccnt 0x0" ::: "memory");
}

__global__ __launch_bounds__(NTHREADS, 1)
void aggx_predictor_kernel(const float* __restrict__ x,
                           const float* __restrict__ t,
                           const float* __restrict__ w,
                           float* __restrict__ out) {
  __shared__ __align__(16) float  w_s[IN_DIM * ROW_STRIDE];     // 266,240 B
  __shared__ __align__(16) float  t_s[SUBS][RPG][OUT_DIM];      //   8,192 B
  __shared__ __align__(16) float  x_s[SUBS][RPG][IN_DIM];       //   8,192 B
  __shared__ __align__(16) float4 xrel4_s[SUBS][IN_DIM];        //   8,192 B

  const int tid  = threadIdx.x;
  const int sub  = tid >> 8;        // which 256-thread sub-block
  const int lane = tid & 255;
  const int b0   = blockIdx.x * ROWS_PER_BLOCK + sub * RPG;

  // ---- Async DMA: w tile + t/x rows straight into LDS (ASYNCcnt path) ----
  // Trip count is exactly 32 and uniform across the block: fully unroll so
  // the wave issues 32 back-to-back async loads with no divergent loop
  // (40 outstanding ASYNCcnt max, well under the 6-bit counter).
  {
    const uint32_t wb = lds_lo32(&w_s[0]);
#pragma unroll
    for (int k = 0; k < CHUNKS_PER_THREAD; ++k) {
      const int c   = tid + k * NTHREADS;
      const int row = c >> 6;     // 64 float4 chunks per w row
      const int q   = c & 63;
      async_ld_b128(wb + (uint32_t)(row * ROW_STRIDE + q * 4) * 4u,
                    w + (size_t)c * 4);
    }
#pragma unroll
    for (int r = 0; r < RPG; ++r) {
      async_ld_b32(lds_lo32(&t_s[sub][r][lane]), t + (size_t)(b0 + r) * OUT_DIM + lane);
      async_ld_b32(lds_lo32(&x_s[sub][r][lane]), x + (size_t)(b0 + r) * IN_DIM  + lane);
    }
    wait_async_zero();            // this wave's async writes landed in LDS
  }
  __syncthreads();                // all waves' data visible

  // ---- Stage 1: x_rel[b,i] = mean_o min(w[i,o], t[b,o]); thread = i, 4 rows/pass ----
  {
    const int i = lane;
    const float4* wrow = reinterpret_cast<const float4*>(&w_s[i * ROW_STRIDE]);
    float a0[RPG], a1[RPG];
#pragma unroll
    for (int r = 0; r < RPG; ++r) { a0[r] = 0.0f; a1[r] = 0.0f; }
#pragma unroll 4
    for (int o4 = 0; o4 < OUT_DIM / 4; ++o4) {
      const float4 wv = wrow[o4];
#pragma unroll
      for (int r = 0; r < RPG; ++r) {
        const float4 tv = reinterpret_cast<const float4*>(t_s[sub][r])[o4]; // broadcast
        a0[r] += fminf(wv.x, tv.x) + fminf(wv.y, tv.y);
        a1[r] += fminf(wv.z, tv.z) + fminf(wv.w, tv.w);
      }
    }
    float4 xr;
    xr.x = (a0[0] + a1[0]) * (1.0f / OUT_DIM);
    xr.y = (a0[1] + a1[1]) * (1.0f / OUT_DIM);
    xr.z = (a0[2] + a1[2]) * (1.0f / OUT_DIM);
    xr.w = (a0[3] + a1[3]) * (1.0f / OUT_DIM);
    xrel4_s[sub][i] = xr;         // rows packed so stage 2 reads one b128
  }
  __syncthreads();

  // ---- Stage 2: i* = argmax_i min(x_rel[b,i], w[i,o]); out = min(x[b,i*], w[i*,o]) ----
  {
    const int o = lane;
    float best[RPG]; int bi[RPG];
#pragma unroll
    for (int r = 0; r < RPG; ++r) { best[r] = -1.0f; bi[r] = 0; }
#pragma unroll 4
    for (int i = 0; i < IN_DIM; ++i) {
      const float  wv = w_s[i * ROW_STRIDE + o];   // conflict-free column read
      const float4 xr = xrel4_s[sub][i];           // broadcast b128
      const float v0 = fminf(xr.x, wv);
      const float v1 = fminf(xr.y, wv);
      const float v2 = fminf(xr.z, wv);
      const float v3 = fminf(xr.w, wv);
      if (v0 > best[0]) { best[0] = v0; bi[0] = i; }   // strict > keeps first max
      if (v1 > best[1]) { best[1] = v1; bi[1] = i; }   // (matches jnp.argmax)
      if (v2 > best[2]) { best[2] = v2; bi[2] = i; }
      if (v3 > best[3]) { best[3] = v3; bi[3] = i; }
    }
#pragma unroll
    for (int r = 0; r < RPG; ++r) {
      const int ii = bi[r];
      out[(size_t)(b0 + r) * OUT_DIM + o] =
          fminf(x_s[sub][r][ii], w_s[ii * ROW_STRIDE + o]);
    }
  }
}

extern "C" void kernel_launch(void* const* d_in, const int* in_sizes, int n_in,
                              void* d_out, int out_size, void* d_ws, size_t ws_size,
                              hipStream_t stream) {
  (void)in_sizes; (void)n_in; (void)out_size; (void)d_ws; (void)ws_size;
  const float* x = (const float*)d_in[0];  // [B, IN]
  const float* t = (const float*)d_in[1];  // [B, OUT]
  const float* w = (const float*)d_in[2];  // [IN, OUT]
  float* out = (float*)d_out;              // [B, OUT]
  aggx_predictor_kernel<<<NBLOCKS, NTHREADS, 0, stream>>>(x, t, w, out);
}